// NeuralEditDistBase_33440615367127
// MI455X (gfx1250) — compile-verified
//
#include <hip/hip_runtime.h>
#include <math.h>

// Problem constants (from reference): B,T,V,C
#define B_ 64
#define T_ 64
#define V_ 64
#define C_ 65

// ---------------- CDNA5 async global->LDS path (guarded) ----------------
#if defined(__has_builtin)
#  if __has_builtin(__builtin_amdgcn_global_load_async_to_lds_b32)
#    define HAVE_ASYNC_LDS 1
#  else
#    define HAVE_ASYNC_LDS 0
#  endif
#  if __has_builtin(__builtin_amdgcn_s_wait_asynccnt)
#    define HAVE_WAIT_ASYNC 1
#  else
#    define HAVE_WAIT_ASYNC 0
#  endif
#else
#  define HAVE_ASYNC_LDS 0
#  define HAVE_WAIT_ASYNC 0
#endif

// Builtin signature (from hipcc diagnostic): first param is addrspace(1) int*,
// second is the LDS destination (addrspace(3) int*), then imm offset, imm cpol.
typedef __attribute__((address_space(1))) int gas_i32;
typedef __attribute__((address_space(3))) int las_i32;

__device__ __forceinline__ void async_copy_f32(const float* g, float* l) {
#if HAVE_ASYNC_LDS
  __builtin_amdgcn_global_load_async_to_lds_b32(
      (gas_i32*)(void*)g, (las_i32*)(void*)l, /*offset=*/0, /*cpol=*/0);
#else
  *l = *g;  // synchronous fallback (still correct)
#endif
}

__device__ __forceinline__ void wait_async_le6() {
#if HAVE_ASYNC_LDS
#  if HAVE_WAIT_ASYNC
  __builtin_amdgcn_s_wait_asynccnt(6);
#  else
  asm volatile("s_wait_asynccnt 6" ::: "memory");
#  endif
#endif
}

__device__ __forceinline__ void wait_async_le0() {
#if HAVE_ASYNC_LDS
#  if HAVE_WAIT_ASYNC
  __builtin_amdgcn_s_wait_asynccnt(0);
#  else
  asm volatile("s_wait_asynccnt 0" ::: "memory");
#  endif
#endif
}

// ---------------- numerics ----------------
__device__ __forceinline__ float lae(float a, float b) {
  // logaddexp with -inf safety (needed because column 0 is a constant map, m=-inf)
  float mx = fmaxf(a, b);
  float mn = fminf(a, b);
  float r = mx + log1pf(__expf(mn - mx));
  return (mx == -__builtin_inff()) ? -__builtin_inff() : r;
}

// Issue the 6 gathered scores of row t (per lane: cols v0,v1) as async LDS copies.
__device__ __forceinline__ void issue_row(const float* __restrict__ S,
                                          const int* __restrict__ subIds,
                                          int idc, int ic0, int ic1,
                                          int t, int v0, int v1,
                                          float* dst /*6 floats in LDS*/) {
  const int ro0 = (t * V_ + v0) * C_;
  const int ro1 = (t * V_ + v1) * C_;
  const int sc0 = subIds[t * V_ + v0];
  const int sc1 = subIds[t * V_ + v1];
  async_copy_f32(S + ro0 + idc, dst + 0);  // del  score, col v0
  async_copy_f32(S + ro1 + idc, dst + 1);  // del  score, col v1
  async_copy_f32(S + ro0 + ic0, dst + 2);  // ins  score, col v0
  async_copy_f32(S + ro1 + ic1, dst + 3);  // ins  score, col v1
  async_copy_f32(S + ro0 + sc0, dst + 4);  // subs score, col v0
  async_copy_f32(S + ro1 + sc1, dst + 5);  // subs score, col v1
}

// One wave32 per batch element. Lane L owns columns v0=2L, v1=2L+1.
// Row update = inclusive wave scan of composed log-affine maps (O(log V) depth).
__global__ __launch_bounds__(32) void neural_edit_dist_kernel(
    const float* __restrict__ scores,   // (B,T,V,C) log-softmax
    const int*   __restrict__ delIds,   // (B,T)
    const int*   __restrict__ insIds,   // (B,V)
    const int*   __restrict__ subIds,   // (B,T,V)
    float*       __restrict__ out) {    // (B,T,V) alpha
  __shared__ float buf[2][32 * 6];

  const int b    = blockIdx.x;
  const int lane = threadIdx.x;          // wave32
  const int v0   = 2 * lane;
  const int v1   = v0 + 1;

  const float* S  = scores + (size_t)b * (T_ * V_ * C_);
  const int*   dI = delIds + b * T_;
  const int*   sI = subIds + b * (T_ * V_);
  float*       O  = out + (size_t)b * (T_ * V_);

  const int ic0 = insIds[b * V_ + v0];
  const int ic1 = insIds[b * V_ + v1];

  // ---- Row 0: alpha0 = cumsum(ins[0,:] with v=0 forced to 0) ----
  const float ir0 = S[v0 * C_ + ic0];
  const float ir1 = S[v1 * C_ + ic1];
  const float x0 = (lane == 0) ? 0.0f : ir0;
  const float x1 = ir1;
  float pairsum = x0 + x1;
  float inc = pairsum;
#pragma unroll
  for (int off = 1; off < 32; off <<= 1) {
    float o = __shfl_up(inc, off, 32);
    if (lane >= off) inc += o;
  }
  const float excl = inc - pairsum;  // sum over columns < v0
  float a0 = excl + x0;              // alpha[0, v0]
  float a1 = a0 + x1;                // alpha[0, v1]
  O[v0] = a0;
  O[v1] = a1;

  // Pre-issue row 1 gathers into LDS double buffer.
  issue_row(S, sI, dI[1], ic0, ic1, 1, v0, v1, &buf[1][lane * 6]);

  for (int t = 1; t < T_; ++t) {
    float* cur = &buf[t & 1][lane * 6];

    if (t + 1 < T_) {
      // Overlap: issue next row's gathers, prefetch t+2's cachelines.
      issue_row(S, sI, dI[t + 1], ic0, ic1, t + 1, v0, v1,
                &buf[(t + 1) & 1][lane * 6]);
      if (t + 2 < T_) {
        __builtin_prefetch(S + ((t + 2) * V_ + v0) * C_, 0, 0);
        __builtin_prefetch(S + ((t + 2) * V_ + v1) * C_, 0, 0);
      }
      // Async loads complete in order: count<=6 means current row's 6 are done.
      wait_async_le6();
    } else {
      wait_async_le0();
    }

    const float d0  = cur[0], d1  = cur[1];
    const float iv0 = cur[2], iv1 = cur[3];
    const float s0  = cur[4], s1  = cur[5];

    // alpha_prev[v0-1] lives in lane-1's second slot.
    const float prevm1 = __shfl_up(a1, 1, 32);

    // Per-column log-affine maps f_v(x) = lae(c, m + x).
    float c1 = lae(d1 + a1, s1 + a0);
    float m1 = iv1;
    float c0, m0;
    if (lane == 0) {  // column 0: constant map (no insertion term)
      c0 = d0 + a0;
      m0 = -__builtin_inff();
    } else {
      c0 = lae(d0 + a0, s0 + prevm1);
      m0 = iv0;
    }

    // Compose the lane's pair: f_{v1} o f_{v0}.
    float M  = m0 + m1;
    float Cv = lae(c1, m1 + c0);

    // Inclusive wave scan of map composition (incoming = lower lanes, applied first):
    //   (M,C) <- (M + Mo, lae(C, M + Co))
#pragma unroll
    for (int off = 1; off < 32; off <<= 1) {
      float Mo = __shfl_up(M, off, 32);
      float Co = __shfl_up(Cv, off, 32);
      if (lane >= off) {
        Cv = lae(Cv, M + Co);
        M  = M + Mo;
      }
    }

    // Column 0's m=-inf makes every prefix a constant map => Cv = alpha[t, v1].
    const float Cexcl = __shfl_up(Cv, 1, 32);  // alpha[t, v0-1]
    const float na0 = (lane == 0) ? c0 : lae(c0, m0 + Cexcl);
    const float na1 = Cv;

    O[t * V_ + v0] = na0;
    O[t * V_ + v1] = na1;
    a0 = na0;
    a1 = na1;
  }
}

extern "C" void kernel_launch(void* const* d_in, const int* in_sizes, int n_in,
                              void* d_out, int out_size, void* d_ws, size_t ws_size,
                              hipStream_t stream) {
  const float* scores = (const float*)d_in[0];   // (B,T,V,C) f32
  const int*   delIds = (const int*)d_in[1];     // (B,T)   i32
  const int*   insIds = (const int*)d_in[2];     // (B,V)   i32
  const int*   subIds = (const int*)d_in[3];     // (B,T,V) i32
  float*       out    = (float*)d_out;           // (B,T,V) f32

  neural_edit_dist_kernel<<<dim3(B_), dim3(32), 0, stream>>>(
      scores, delIds, insIds, subIds, out);
}